// PPGCN_14688788152762
// MI455X (gfx1250) — compile-verified
//
#include <hip/hip_runtime.h>

typedef float v2f __attribute__((ext_vector_type(2)));
typedef float v8f __attribute__((ext_vector_type(8)));

#define NN   50000      // nodes
#define BB   2048       // batch nodes
#define INF  256        // input features
#define HID  512        // hidden
#define OUTF 256        // output features
#define EE   1000000    // edges per relation
#define RR   3          // relations
#define BN_EPS 1e-5f

// ---------------- utility fills ----------------
__global__ void fill_i32(int* p, int n, int v) {
  int i = blockIdx.x * blockDim.x + threadIdx.x;
  if (i < n) p[i] = v;
}
__global__ void fill_f32(float* p, int n, float v) {
  int i = blockIdx.x * blockDim.x + threadIdx.x;
  if (i < n) p[i] = v;
}
__global__ void set_in_batch(const long long* __restrict__ bn, int* __restrict__ ib) {
  int i = blockIdx.x * blockDim.x + threadIdx.x;
  if (i < BB) ib[(int)bn[i]] = 1;
}

// ---------------- graph preprocessing ----------------
__global__ void mark_appear(const long long* __restrict__ src, const long long* __restrict__ dst,
                            const int* __restrict__ ib, int* __restrict__ ap) {
  int e = blockIdx.x * blockDim.x + threadIdx.x;
  if (e >= EE) return;
  int s = (int)src[e], d = (int)dst[e];
  if (ib[s] && ib[d]) { ap[s] = 1; ap[d] = 1; }
}

// single-workgroup tiled inclusive scan over NN flags: new_id = cumsum(appear)-1
__global__ void scan_new_id(const int* __restrict__ ap, int* __restrict__ nid) {
  __shared__ int tile[1024];
  __shared__ int carry;
  if (threadIdx.x == 0) carry = 0;
  __syncthreads();
  for (int base = 0; base < NN; base += 1024) {
    int i = base + threadIdx.x;
    int v = (i < NN) ? ap[i] : 0;
    tile[threadIdx.x] = v;
    __syncthreads();
    for (int o = 1; o < 1024; o <<= 1) {
      int t = (threadIdx.x >= o) ? tile[threadIdx.x - o] : 0;
      __syncthreads();
      tile[threadIdx.x] += t;
      __syncthreads();
    }
    int inc = tile[threadIdx.x] + carry;
    if (i < NN) nid[i] = inc - 1;
    __syncthreads();
    if (threadIdx.x == 1023) carry = inc;
    __syncthreads();
  }
}

__global__ void accum_deg(const long long* __restrict__ src, const long long* __restrict__ dst,
                          const int* __restrict__ ib, const int* __restrict__ nid,
                          float* __restrict__ deg) {
  int e = blockIdx.x * blockDim.x + threadIdx.x;
  if (e >= EE) return;
  int s = (int)src[e], d = (int)dst[e];
  if (ib[s] && ib[d]) atomicAdd(&deg[nid[d]], 1.0f);
}

__global__ void make_dinv(float* deg) {
  int i = blockIdx.x * blockDim.x + threadIdx.x;
  if (i < BB) deg[i] = rsqrtf(deg[i]);
}

// ---------------- WMMA fp32 GEMM (optionally row-gathered A) ----------------
// out[M x Nout] = A[gathered rows, K] @ W[K x Nout]
// Workgroup = 256 threads = 8 waves; wave (mw,ng) computes rows [blk.x*64+mw*16, +16)
// x cols [blk.y*128+ng*64, +64) as 4 WMMA 16x16 tiles sharing one A fragment.
__global__ __launch_bounds__(256)
void gemm_wmma(const float* __restrict__ A, const long long* __restrict__ gather, int lda,
               const float* __restrict__ W, int K, int Nout, float* __restrict__ out) {
  int lane = threadIdx.x & 31;
  int wave = threadIdx.x >> 5;
  int mw = wave & 3, ng = wave >> 2;
  int row0 = blockIdx.x * 64 + mw * 16;
  int col0 = blockIdx.y * 128 + ng * 64;
  int lm = lane & 15, half = lane >> 4;

  size_t rowA = gather ? (size_t)gather[row0 + lm] : (size_t)(row0 + lm);
  const float* aptr = A + rowA * (size_t)lda;

  v8f acc0 = {}, acc1 = {}, acc2 = {}, acc3 = {};
  for (int k0 = 0; k0 < K; k0 += 4) {
    // A 16x4 f32 fragment: lanes 0-15 hold K=k0,k0+1 ; lanes 16-31 hold K=k0+2,k0+3
    v2f a = *(const v2f*)(aptr + k0 + 2 * half);
    const float* wp = W + (size_t)(k0 + 2 * half) * Nout;
    int n0 = col0 + lm;
    v2f b0, b1v, b2v, b3;
    b0.x  = wp[n0];       b0.y  = wp[Nout + n0];
    b1v.x = wp[n0 + 16];  b1v.y = wp[Nout + n0 + 16];
    b2v.x = wp[n0 + 32];  b2v.y = wp[Nout + n0 + 32];
    b3.x  = wp[n0 + 48];  b3.y  = wp[Nout + n0 + 48];
    acc0 = __builtin_amdgcn_wmma_f32_16x16x4_f32(false, a, false, b0,  (short)0, acc0, false, false);
    acc1 = __builtin_amdgcn_wmma_f32_16x16x4_f32(false, a, false, b1v, (short)0, acc1, false, false);
    acc2 = __builtin_amdgcn_wmma_f32_16x16x4_f32(false, a, false, b2v, (short)0, acc2, false, false);
    acc3 = __builtin_amdgcn_wmma_f32_16x16x4_f32(false, a, false, b3,  (short)0, acc3, false, false);
  }
  v8f accs[4] = {acc0, acc1, acc2, acc3};
  for (int t = 0; t < 4; t++) {
    int n = col0 + 16 * t + lm;
#pragma unroll
    for (int j = 0; j < 8; j++)
      out[(size_t)(row0 + j + 8 * half) * Nout + n] = accs[t][j];
  }
}

// ---------------- GCN pieces ----------------
// agg[i][c] = dinv[i]^2 * xw[i][c] + bias[c]
__global__ void self_bias(const float* __restrict__ xw, const float* __restrict__ dinv,
                          const float* __restrict__ bias, float* __restrict__ agg, int F) {
  int idx = blockIdx.x * blockDim.x + threadIdx.x;
  if (idx >= BB * F) return;
  int i = idx / F, c = idx - i * F;
  float di = dinv[i];
  agg[idx] = di * di * xw[idx] + bias[c];
}

// sparse scatter-add over kept edges: agg[d] += dinv[s]*dinv[d] * xw[s]
__global__ __launch_bounds__(256)
void edge_agg(const long long* __restrict__ src, const long long* __restrict__ dst,
              const int* __restrict__ ib, const int* __restrict__ nid,
              const float* __restrict__ dinv, const float* __restrict__ xw,
              float* __restrict__ agg, int F) {
  for (int e = blockIdx.x; e < EE; e += gridDim.x) {
    int s = (int)src[e], d = (int)dst[e];
    if (!(ib[s] && ib[d])) continue;
    int sn = nid[s], dn = nid[d];
    float norm = dinv[sn] * dinv[dn];
    const float* xs = xw + (size_t)sn * F;
    float* ad = agg + (size_t)dn * F;
    for (int c = threadIdx.x; c < F; c += blockDim.x)
      atomicAdd(&ad[c], norm * xs[c]);
  }
}

// per-column partial sums (coalesced: consecutive threads -> consecutive columns)
__global__ __launch_bounds__(256)
void bn_stats(const float* __restrict__ h, float* __restrict__ cs, float* __restrict__ cq) {
  float s0 = 0.f, s1 = 0.f, q0 = 0.f, q1 = 0.f;
  int r0 = blockIdx.x * 64;
  for (int r = r0; r < r0 + 64; r++) {
    float v0 = h[(size_t)r * HID + threadIdx.x];
    float v1 = h[(size_t)r * HID + threadIdx.x + 256];
    s0 += v0; q0 += v0 * v0;
    s1 += v1; q1 += v1 * v1;
  }
  atomicAdd(&cs[threadIdx.x],       s0);
  atomicAdd(&cq[threadIdx.x],       q0);
  atomicAdd(&cs[threadIdx.x + 256], s1);
  atomicAdd(&cq[threadIdx.x + 256], q1);
}

__global__ void bn_apply(float* __restrict__ h, const float* __restrict__ cs,
                         const float* __restrict__ cq, const float* __restrict__ gamma,
                         const float* __restrict__ beta) {
  int idx = blockIdx.x * blockDim.x + threadIdx.x;
  if (idx >= BB * HID) return;
  int c = idx % HID;
  float mu  = cs[c] * (1.0f / BB);
  float var = cq[c] * (1.0f / BB) - mu * mu;   // biased, like jnp.var
  h[idx] = (h[idx] - mu) * rsqrtf(var + BN_EPS) * gamma[c] + beta[c];
}

// ---------------- launcher ----------------
extern "C" void kernel_launch(void* const* d_in, const int* in_sizes, int n_in,
                              void* d_out, int out_size, void* d_ws, size_t ws_size,
                              hipStream_t stream) {
  (void)in_sizes; (void)n_in; (void)out_size; (void)ws_size;
  const float*     features    = (const float*)d_in[0];
  const float*     W1          = (const float*)d_in[1];
  const float*     b1          = (const float*)d_in[2];
  const float*     W2          = (const float*)d_in[3];
  const float*     b2          = (const float*)d_in[4];
  const float*     gamma       = (const float*)d_in[5];
  const float*     beta        = (const float*)d_in[6];
  const long long* edge_index  = (const long long*)d_in[7];
  const long long* batch_nodes = (const long long*)d_in[8];
  float* out = (float*)d_out;

  char* ws = (char*)d_ws;
  size_t off = 0;
  auto alloc = [&](size_t bytes) -> void* {
    void* p = ws + off;
    off = (off + bytes + 255) & ~(size_t)255;
    return p;
  };
  int*   ib     = (int*)alloc((size_t)NN * 4);
  int*   ap     = (int*)alloc((size_t)NN * 4);
  int*   nid    = (int*)alloc((size_t)NN * 4);
  float* dinv   = (float*)alloc((size_t)BB * 4);
  float* xw     = (float*)alloc((size_t)BB * HID * 4);   // reused for both layers
  float* agg1   = (float*)alloc((size_t)BB * HID * 4);
  float* colsum = (float*)alloc((size_t)HID * 4);
  float* colsq  = (float*)alloc((size_t)HID * 4);

  const int T = 256;
  fill_i32<<<(NN + T - 1) / T, T, 0, stream>>>(ib, NN, 0);
  set_in_batch<<<(BB + T - 1) / T, T, 0, stream>>>(batch_nodes, ib);

  for (int r = 0; r < RR; r++) {
    const long long* src = edge_index + (size_t)r * 2 * EE;
    const long long* dst = src + EE;
    const float* feat_r  = features + (size_t)r * NN * INF;
    float* outr          = out + (size_t)r * BB * OUTF;

    // renumbering + degrees
    fill_i32<<<(NN + T - 1) / T, T, 0, stream>>>(ap, NN, 0);
    mark_appear<<<(EE + T - 1) / T, T, 0, stream>>>(src, dst, ib, ap);
    scan_new_id<<<1, 1024, 0, stream>>>(ap, nid);
    fill_f32<<<(BB + T - 1) / T, T, 0, stream>>>(dinv, BB, 1.0f);   // +1 self loop
    accum_deg<<<(EE + T - 1) / T, T, 0, stream>>>(src, dst, ib, nid, dinv);
    make_dinv<<<(BB + T - 1) / T, T, 0, stream>>>(dinv);

    // layer 1: gathered X @ W1, self-loop+bias, sparse aggregation
    dim3 g1(BB / 64, HID / 128);
    gemm_wmma<<<g1, 256, 0, stream>>>(feat_r, batch_nodes, INF, W1, INF, HID, xw);
    self_bias<<<(BB * HID + T - 1) / T, T, 0, stream>>>(xw, dinv, b1, agg1, HID);
    edge_agg<<<4096, T, 0, stream>>>(src, dst, ib, nid, dinv, xw, agg1, HID);

    // batchnorm (training-mode batch stats), in place
    fill_f32<<<(HID + T - 1) / T, T, 0, stream>>>(colsum, HID, 0.0f);
    fill_f32<<<(HID + T - 1) / T, T, 0, stream>>>(colsq,  HID, 0.0f);
    bn_stats<<<BB / 64, 256, 0, stream>>>(agg1, colsum, colsq);
    bn_apply<<<(BB * HID + T - 1) / T, T, 0, stream>>>(agg1, colsum, colsq, gamma, beta);

    // layer 2: H @ W2 straight into d_out block for this relation
    dim3 g2(BB / 64, OUTF / 128);
    gemm_wmma<<<g2, 256, 0, stream>>>(agg1, nullptr, HID, W2, HID, OUTF, xw);
    self_bias<<<(BB * OUTF + T - 1) / T, T, 0, stream>>>(xw, dinv, b2, outr, OUTF);
    edge_agg<<<4096, T, 0, stream>>>(src, dst, ib, nid, dinv, xw, outr, OUTF);
  }
}